// L2LBGRLGConv_68539088109711
// MI455X (gfx1250) — compile-verified
//
#include <hip/hip_runtime.h>
#include <hip/hip_bf16.h>
#include <cstdint>
#include <cstddef>

#define NEG_SLOPE 0.2f
#define BN_EPS 1e-5f

typedef __attribute__((ext_vector_type(2))) float v2f;
typedef __attribute__((ext_vector_type(8))) float v8f;

// Order-preserving int encoding of float for atomicMax (handles negatives).
__device__ __forceinline__ int fenc_f32(float f) {
  int i = __float_as_int(f);
  return i >= 0 ? i : (i ^ 0x7fffffff);
}
__device__ __forceinline__ float fdec_f32(int i) {
  return __int_as_float(i >= 0 ? i : (i ^ 0x7fffffff));
}
// fenc(-inf) = 0xff800000 ^ 0x7fffffff = 0x807fffff
#define FENC_NEG_INF 0x807fffff

// ---------------------------------------------------------------------------
// GEMM: Y[nrows, C] = X[nrows, K] @ W[C, K]^T (+ optional bias[C])
// One wave32 per 16x16 output tile, V_WMMA_F32_16X16X4_F32, K-step of 4.
// A 16x4 f32 frag: lanes 0-15 hold M=lane, K=k0..k0+1; lanes 16-31 K=k0+2..k0+3
// B 4x16 f32 frag: lanes 0-15 hold N=lane, K=k0..k0+1; lanes 16-31 K=k0+2..k0+3
// C/D 16x16 f32:   VGPR v -> M = v + 8*(lane>=16), N = lane&15
// ---------------------------------------------------------------------------
__global__ __launch_bounds__(256) void gemm_wmma_f32(
    const float* __restrict__ X, const float* __restrict__ W,
    const float* __restrict__ bias, float* __restrict__ Y,
    int nrows, int K, int C) {
  const int lane = threadIdx.x & 31;
  const int wave = threadIdx.x >> 5;
  const int colTiles = C >> 4;
  const int rowTiles = (nrows + 15) >> 4;
  const int tile = blockIdx.x * 8 + wave;
  if (tile >= rowTiles * colTiles) return;  // wave-uniform -> EXEC all ones below
  const int rt = tile / colTiles;
  const int ct = tile - rt * colTiles;

  const int lo = lane & 15;
  const int kofs = (lane >> 4) << 1;  // 0 or 2
  const float2* __restrict__ A2 =
      reinterpret_cast<const float2*>(X + (size_t)(rt * 16 + lo) * K + kofs);
  const float2* __restrict__ B2 =
      reinterpret_cast<const float2*>(W + (size_t)(ct * 16 + lo) * K + kofs);

  v8f acc = {};
  const int ksteps = K >> 2;
#pragma unroll 4
  for (int kk = 0; kk < ksteps; ++kk) {
    const float2 af = A2[kk * 2];
    const float2 bf = B2[kk * 2];
    v2f a; a[0] = af.x; a[1] = af.y;
    v2f b; b[0] = bf.x; b[1] = bf.y;
    // 8 args: (neg_a, A, neg_b, B, c_mod, C, reuse_a, reuse_b)
    acc = __builtin_amdgcn_wmma_f32_16x16x4_f32(false, a, false, b,
                                                (short)0, acc, false, false);
  }

  const int mbase = rt * 16 + ((lane >> 4) << 3);
  const int n = ct * 16 + lo;
  const float bv = bias ? bias[n] : 0.0f;
#pragma unroll
  for (int v = 0; v < 8; ++v) {
    Y[(size_t)(mbase + v) * C + n] = acc[v] + bv;
  }
}

// ---------------------------------------------------------------------------
// Edge-phase kernels. Edges 0..E-1 come from edge_index; edges E..E+N-1 are
// self loops (src = dst = e - E). One wave per task = (edge, head).
// ---------------------------------------------------------------------------
__global__ __launch_bounds__(256) void gat_edge_logits(
    const float* __restrict__ xl, const float* __restrict__ xr,
    const float* __restrict__ att, const int* __restrict__ ei,
    int E, int Etot, float* __restrict__ logit) {
  const int task = blockIdx.x * 8 + (threadIdx.x >> 5);
  if (task >= Etot * 2) return;
  const int lane = threadIdx.x & 31;
  const int e = task >> 1, h = task & 1;
  int s, d;
  if (e < E) { s = ei[e]; d = ei[E + e]; } else { s = d = e - E; }
  const int c = lane << 1;
  const float2 xlv = *reinterpret_cast<const float2*>(xl + (size_t)s * 128 + h * 64 + c);
  const float2 xrv = *reinterpret_cast<const float2*>(xr + (size_t)d * 128 + h * 64 + c);
  const float2 at  = *reinterpret_cast<const float2*>(att + h * 64 + c);
  float ex = xlv.x + xrv.x; ex = ex >= 0.0f ? ex : NEG_SLOPE * ex;
  float ey = xlv.y + xrv.y; ey = ey >= 0.0f ? ey : NEG_SLOPE * ey;
  float p = ex * at.x + ey * at.y;
#pragma unroll
  for (int off = 16; off > 0; off >>= 1) p += __shfl_xor(p, off, 32);
  if (lane == 0) logit[task] = p;
}

__global__ __launch_bounds__(256) void gat_segment_max(
    const float* __restrict__ logit, const int* __restrict__ ei,
    int E, int Etot, int N, int* __restrict__ menc) {
  const int idx = blockIdx.x * 256 + threadIdx.x;
  if (idx >= Etot * 2) return;
  const int e = idx >> 1, h = idx & 1;
  const int d = (e < E) ? ei[E + e] : (e - E);
  atomicMax(&menc[d * 2 + h], fenc_f32(logit[idx]));
}

__global__ __launch_bounds__(256) void gat_exp_sum(
    const float* __restrict__ logit, const int* __restrict__ menc,
    const int* __restrict__ ei, int E, int Etot,
    float* __restrict__ aexp, float* __restrict__ ssum) {
  const int idx = blockIdx.x * 256 + threadIdx.x;
  if (idx >= Etot * 2) return;
  const int e = idx >> 1, h = idx & 1;
  const int d = (e < E) ? ei[E + e] : (e - E);
  const float mv = fdec_f32(menc[d * 2 + h]);
  const float av = expf(logit[idx] - mv);
  aexp[idx] = av;
  atomicAdd(&ssum[d * 2 + h], av);
}

__global__ __launch_bounds__(256) void gat_aggregate(
    const float* __restrict__ xl, const float* __restrict__ aexp,
    const float* __restrict__ ssum, const int* __restrict__ ei,
    int E, int Etot, float* __restrict__ agg) {
  const int task = blockIdx.x * 8 + (threadIdx.x >> 5);
  if (task >= Etot * 2) return;
  const int lane = threadIdx.x & 31;
  const int e = task >> 1, h = task & 1;
  int s, d;
  if (e < E) { s = ei[e]; d = ei[E + e]; } else { s = d = e - E; }
  const float alpha = aexp[task] / ssum[d * 2 + h];
  const int c = lane << 1;
  const float2 xlv = *reinterpret_cast<const float2*>(xl + (size_t)s * 128 + h * 64 + c);
  float* dst = agg + (size_t)d * 128 + h * 64 + c;
  atomicAdd(dst,     alpha * xlv.x);
  atomicAdd(dst + 1, alpha * xlv.y);
}

// head mean + bias + PReLU
__global__ __launch_bounds__(256) void gat_finalize(
    const float* __restrict__ agg, const float* __restrict__ b,
    const float* __restrict__ actw, float* __restrict__ z, int total) {
  const int idx = blockIdx.x * 256 + threadIdx.x;
  if (idx >= total) return;
  const int n = idx >> 6, c = idx & 63;
  float v = 0.5f * (agg[(size_t)n * 128 + c] + agg[(size_t)n * 128 + 64 + c]) + b[c];
  const float w = actw[0];
  z[idx] = v >= 0.0f ? v : w * v;
}

// ---------------------------------------------------------------------------
// BatchNorm (training mode, biased variance)
// ---------------------------------------------------------------------------
__global__ __launch_bounds__(256) void bn_stats(
    const float* __restrict__ x, int n, float* __restrict__ mu_rstd) {
  const int c = blockIdx.x;  // 64 blocks, one per channel
  float sum = 0.0f, sq = 0.0f;
  for (int i = threadIdx.x; i < n; i += 256) {
    const float v = x[(size_t)i * 64 + c];
    sum += v; sq += v * v;
  }
  __shared__ float ssum[256], ssq[256];
  ssum[threadIdx.x] = sum; ssq[threadIdx.x] = sq;
  __syncthreads();
  for (int st = 128; st > 0; st >>= 1) {
    if (threadIdx.x < st) {
      ssum[threadIdx.x] += ssum[threadIdx.x + st];
      ssq[threadIdx.x]  += ssq[threadIdx.x + st];
    }
    __syncthreads();
  }
  if (threadIdx.x == 0) {
    const float mu = ssum[0] / (float)n;
    const float var = ssq[0] / (float)n - mu * mu;
    mu_rstd[c] = mu;
    mu_rstd[64 + c] = rsqrtf(var + BN_EPS);
  }
}

__global__ __launch_bounds__(256) void bn_apply(
    const float* __restrict__ x, const float* __restrict__ mu_rstd,
    const float* __restrict__ g, const float* __restrict__ bta,
    const float* __restrict__ prelu_w, float* __restrict__ y, int total) {
  const int idx = blockIdx.x * 256 + threadIdx.x;
  if (idx >= total) return;
  const int c = idx & 63;
  float v = (x[idx] - mu_rstd[c]) * mu_rstd[64 + c] * g[c] + bta[c];
  if (prelu_w) { const float w = prelu_w[0]; v = v >= 0.0f ? v : w * v; }
  y[idx] = v;
}

__global__ __launch_bounds__(256) void fill_i32(int* __restrict__ p, int v, int n) {
  const int i = blockIdx.x * 256 + threadIdx.x;
  if (i < n) p[i] = v;
}

// ---------------------------------------------------------------------------
extern "C" void kernel_launch(void* const* d_in, const int* in_sizes, int n_in,
                              void* d_out, int out_size, void* d_ws, size_t ws_size,
                              hipStream_t stream) {
  (void)n_in; (void)out_size; (void)ws_size;
  const float* x    = (const float*)d_in[0];
  const int*   ei   = (const int*)d_in[1];
  const float* Wl1  = (const float*)d_in[2];
  const float* Wr1  = (const float*)d_in[3];
  const float* att1 = (const float*)d_in[4];
  const float* b1   = (const float*)d_in[5];
  const float* Wl2  = (const float*)d_in[6];
  const float* Wr2  = (const float*)d_in[7];
  const float* att2 = (const float*)d_in[8];
  const float* b2   = (const float*)d_in[9];
  const float* actw = (const float*)d_in[10];
  const float* bng  = (const float*)d_in[11];
  const float* bnb  = (const float*)d_in[12];
  const float* pW   = (const float*)d_in[13];
  const float* pb   = (const float*)d_in[14];
  const float* pbng = (const float*)d_in[15];
  const float* pbnb = (const float*)d_in[16];
  const float* pactw= (const float*)d_in[17];

  const int N    = in_sizes[0] / 256;
  const int E    = in_sizes[1] / 2;
  const int Etot = E + N;

  // workspace layout (floats)
  float* f = (float*)d_ws;
  size_t o = 0;
  float* xl    = f + o; o += (size_t)N * 128;
  float* xr    = f + o; o += (size_t)N * 128;
  float* agg   = f + o; o += (size_t)N * 128;
  float* logit = f + o; o += (size_t)Etot * 2;
  float* aexp  = f + o; o += (size_t)Etot * 2;
  float* ssum  = f + o; o += (size_t)N * 2;
  int*   menc  = (int*)(f + o); o += (size_t)N * 2;
  float* z1    = f + o; o += (size_t)N * 64;
  float* z2    = f + o; o += (size_t)N * 64;
  float* praw  = f + o; o += (size_t)N * 64;
  float* stats = f + o; o += 128;

  float* z_out = (float*)d_out;             // [N,64]
  float* p_out = (float*)d_out + (size_t)N * 64;  // [N,64]

  const int rowTiles = (N + 15) >> 4;
  const int tasksEH  = Etot * 2;
  const dim3 blk(256);

  auto run_layer = [&](const float* zin, int K, const float* Wl, const float* Wr,
                       const float* att, const float* bvec, float* zout) {
    const int gemmBlocks = (rowTiles * 8 + 7) / 8;  // C = 128 -> 8 col tiles
    gemm_wmma_f32<<<gemmBlocks, blk, 0, stream>>>(zin, Wl, nullptr, xl, N, K, 128);
    gemm_wmma_f32<<<gemmBlocks, blk, 0, stream>>>(zin, Wr, nullptr, xr, N, K, 128);
    hipMemsetAsync(ssum, 0, (size_t)N * 2 * sizeof(float), stream);
    hipMemsetAsync(agg, 0, (size_t)N * 128 * sizeof(float), stream);
    fill_i32<<<(N * 2 + 255) / 256, blk, 0, stream>>>(menc, FENC_NEG_INF, N * 2);
    gat_edge_logits<<<(tasksEH + 7) / 8, blk, 0, stream>>>(xl, xr, att, ei, E, Etot, logit);
    gat_segment_max<<<(tasksEH + 255) / 256, blk, 0, stream>>>(logit, ei, E, Etot, N, menc);
    gat_exp_sum<<<(tasksEH + 255) / 256, blk, 0, stream>>>(logit, menc, ei, E, Etot, aexp, ssum);
    gat_aggregate<<<(tasksEH + 7) / 8, blk, 0, stream>>>(xl, aexp, ssum, ei, E, Etot, agg);
    gat_finalize<<<(N * 64 + 255) / 256, blk, 0, stream>>>(agg, bvec, actw, zout, N * 64);
  };

  // encoder
  run_layer(x, 256, Wl1, Wr1, att1, b1, z1);   // layer 1 + PReLU
  run_layer(z1, 64, Wl2, Wr2, att2, b2, z2);   // layer 2 + PReLU
  bn_stats<<<64, blk, 0, stream>>>(z2, N, stats);
  bn_apply<<<(N * 64 + 255) / 256, blk, 0, stream>>>(z2, stats, bng, bnb, nullptr,
                                                     z_out, N * 64);
  // projection head: Linear(+bias) -> BN -> PReLU
  const int projBlocks = (rowTiles * 4 + 7) / 8;  // C = 64 -> 4 col tiles
  gemm_wmma_f32<<<projBlocks, blk, 0, stream>>>(z_out, pW, pb, praw, N, 64, 64);
  bn_stats<<<64, blk, 0, stream>>>(praw, N, stats);
  bn_apply<<<(N * 64 + 255) / 256, blk, 0, stream>>>(praw, stats, pbng, pbnb, pactw,
                                                     p_out, N * 64);
}